// ConvCaps_79602923864485
// MI455X (gfx1250) — compile-verified
//
#include <hip/hip_runtime.h>
#include <hip/hip_bf16.h>
#include <math.h>
#include <stdint.h>

// ---------------------------------------------------------------------------
// ConvCaps (EM routing) forward for MI455X / gfx1250, wave32.
//
//   A=32 in-caps, B=32 out-caps, K=3, P=4, stride=2, H=W=16 -> oh=ow=7, l=49
//   KKA = 288, PSIZE = 16, batch b = 8, bl = b*l = 392
//
// Kernel 1: pose transform as 288 batched GEMMs (1568 x 4) @ (4 x 128) using
//           V_WMMA_F32_16X16X4_F32 (K=4 is the native contraction depth).
//           One wave per (row-tile, n); the A fragment stays in VGPRs across
//           all 8 column tiles (8 WMMAs per wave).
// Kernel 2: EM routing, one workgroup per (b,l), r/a_u/mu/sigma in LDS,
//           v streamed from the workspace (bandwidth-bound: ~5 passes over
//           231 MB ~= 50 us at 23.3 TB/s). a_u gathered via
//           GLOBAL_LOAD_ASYNC_TO_LDS_B32 (ASYNCcnt path).
// ---------------------------------------------------------------------------

typedef float v2f __attribute__((ext_vector_type(2)));
typedef float v8f __attribute__((ext_vector_type(8)));

#define CAPS_A   32
#define CAPS_B   32
#define KKA      288          // 9 * 32, n = kk*32 + a  (KK-major per reference)
#define NL       49           // 7*7
#define NBL      392          // 8 * 49
#define NROWS    1568         // NBL * 4   (rows of the batched GEMM: bl*4 + i)
#define NCOLS    128          // 32 * 4    (cols: o*4 + j)
#define VSTRIDE_N 200704      // NROWS * NCOLS, per-n slab of v in the workspace
#define EPSF     1e-12f
#define LN2PI_F  1.8378770664093453f

// ---------------------------------------------------------------------------
// Kernel 1: v[n, row=bl*4+i, col=o*4+j] = sum_m pose_u[bl,n,i,m] * W[n,o,m,j]
// One wave32 per (rt, n); loops the 8 column tiles with A resident in VGPRs.
// EXEC is all ones as WMMA requires.
// ---------------------------------------------------------------------------
__global__ __launch_bounds__(32)
void convcaps_pose_wmma(const float* __restrict__ pose,   // (8, 512, 16, 16)
                        const float* __restrict__ W,      // (288, 32, 4, 4)
                        float* __restrict__ v_ws)         // (288, 1568, 128)
{
  const int rt   = blockIdx.x;          // row tile 0..97
  const int n    = blockIdx.y;          // 0..287
  const int lane = threadIdx.x;         // 0..31
  const int half = lane >> 4;           // half-wave selector
  const int l15  = lane & 15;

  // decode n = kk*32 + a ; kk = ki*3 + kj
  const int kk = n >> 5;
  const int a  = n & 31;
  const int ki = kk / 3, kj = kk % 3;

  // ---- A fragment: 16x4 f32 (2 VGPRs), loaded ONCE, reused for 8 tiles ----
  // ISA layout: lanes 0-15 -> {K=0 in reg0, K=1 in reg1}, lanes 16-31 -> {K=2, K=3}
  const int g  = rt * 16 + l15;         // global row = bl*4 + i
  const int bl = g >> 2;
  const int i  = g & 3;
  const int b  = bl / NL;
  const int l  = bl % NL;
  const int oy = l / 7, ox = l % 7;
  const int py = ki + 2 * oy;           // unfold: input pixel
  const int px = kj + 2 * ox;
  const int m0 = half * 2;
  // pose channel = a*16 + i*4 + m ; channel stride = 256 (16*16)
  const float* pbase =
      pose + (size_t)(b * 512 + a * 16 + i * 4) * 256 + py * 16 + px;
  v2f afrag;
  afrag.x = pbase[(size_t)(m0 + 0) * 256];
  afrag.y = pbase[(size_t)(m0 + 1) * 256];

  // ---- B fragment base (ct = 0): 4x16 f32, row-striped across lanes -------
  // reg0: K = {0 | 2}, reg1: K = {1 | 3} per half-wave; N = lane&15
  // col c = ct*16 + l15 -> o = c>>2, j = c&3 ; addr = n*512 + o*16 + j + m*4
  // ct -> ct+1 advances the address by exactly 64 floats.
  const int o0 = l15 >> 2, j0 = l15 & 3;
  const float* wbase = W + (size_t)n * 512 + o0 * 16 + j0 + m0 * 4;

  // ---- D store base (ct = 0): row (rt*16 + r + half*8), col (ct*16 + l15) -
  float* dst0 = v_ws + (size_t)n * VSTRIDE_N +
                (size_t)(rt * 16 + half * 8) * NCOLS + l15;

#pragma unroll
  for (int ct = 0; ct < 8; ++ct) {
    v2f bfrag;
    bfrag.x = wbase[ct * 64 + 0];
    bfrag.y = wbase[ct * 64 + 4];

    v8f acc = {};
    acc = __builtin_amdgcn_wmma_f32_16x16x4_f32(
        /*neg_a=*/false, afrag, /*neg_b=*/false, bfrag,
        /*c_mod=*/(short)0, acc, /*reuse_a=*/false, /*reuse_b=*/false);

    float* dst = dst0 + ct * 16;
#pragma unroll
    for (int r = 0; r < 8; ++r)
      dst[(size_t)r * NCOLS] = acc[r];
  }
}

// ---------------------------------------------------------------------------
// Kernel 2: EM routing. One 512-thread block per (b,l). LDS ~44 KB.
// thread t -> (o = t>>4, ij = t&15) for the mu/sigma phase,
//             (row = t>>5, o = t&31) for the r-update phase.
// ---------------------------------------------------------------------------
__global__ __launch_bounds__(512)
void convcaps_routing(const float* __restrict__ a_in,    // (8, 32, 16, 16)
                      const float* __restrict__ v_ws,    // (288, 1568, 128)
                      const float* __restrict__ beta_u,  // (32)
                      const float* __restrict__ beta_a,  // (32)
                      float* __restrict__ out_a,         // (8, 32, 7, 7)
                      float* __restrict__ out_pose)      // (8, 512, 7, 7)
{
  __shared__ float sh_r[KKA * CAPS_B];     // 36864 B
  __shared__ float sh_au[KKA];
  __shared__ float sh_mu[CAPS_B * 16];
  __shared__ float sh_ss[CAPS_B * 16];
  __shared__ float sh_rsum[CAPS_B];
  __shared__ float sh_logterm[CAPS_B];
  __shared__ float sh_la[CAPS_B];
  __shared__ float sh_aout[CAPS_B];
  __shared__ float sh_lnap[16 * CAPS_B];

  const int bl  = blockIdx.x;              // 0..391
  const int b   = bl / NL;
  const int l   = bl % NL;
  const int oy  = l / 7, ox = l % 7;
  const int tid = threadIdx.x;

  // r init to 1
  for (int idx = tid; idx < KKA * CAPS_B; idx += 512) sh_r[idx] = 1.0f;

  // a_u gather (unfold of a_in): per-lane async copy, memory -> LDS, no VGPR
  // round trip.  Waves 0..8 are fully active (288 = 9*32) so EXEC is dense.
  if (tid < KKA) {
    const int n  = tid;
    const int kk = n >> 5, a = n & 31;
    const int ki = kk / 3, kj = kk % 3;
    const uint64_t gaddr = (uint64_t)(uintptr_t)
        &a_in[((b * 32 + a) * 16 + (ki + 2 * oy)) * 16 + (kj + 2 * ox)];
    const uint32_t ldsoff = (uint32_t)(uintptr_t)&sh_au[n];
    asm volatile("global_load_async_to_lds_b32 %0, %1, off"
                 :: "v"(ldsoff), "v"(gaddr)
                 : "memory");
  }
  asm volatile("s_wait_asynccnt 0x0" ::: "memory");
  __syncthreads();

  const int o  = tid >> 4;                 // 0..31
  const int ij = tid & 15;                 // i*4 + j
  const int ii = ij >> 2, jj = ij & 3;
  const float* vp = v_ws + (size_t)(bl * 4 + ii) * NCOLS + o * 4 + jj;

  const float lam_tab[3] = {0.0005f, 0.000975f, 0.00142625f}; // 0.01*(1-0.95^(i+1))

  for (int it = 0; it < 3; ++it) {
    const float lam = lam_tab[it];

    // --- r_sum[o] = sum_n r[n][o] * a_u[n] -------------------------------
    if (tid < CAPS_B) {
      float rs = 0.f;
      for (int n = 0; n < KKA; ++n) rs += sh_r[n * CAPS_B + tid] * sh_au[n];
      sh_rsum[tid] = rs;
    }
    __syncthreads();

    // --- one-pass mu / sigma over n (v streamed, prefetch ahead) ---------
    {
      const float inv = 1.0f / (sh_rsum[o] + EPSF);
      float s1 = 0.f, s2 = 0.f;
      for (int n = 0; n < KKA; ++n) {
        if (n + 16 < KKA)
          __builtin_prefetch(vp + (size_t)(n + 16) * VSTRIDE_N, 0, 0);
        const float cc = sh_r[n * CAPS_B + o] * sh_au[n] * inv;
        const float vv = vp[(size_t)n * VSTRIDE_N];
        s1 = fmaf(cc, vv, s1);
        s2 = fmaf(cc * vv, vv, s2);
      }
      sh_mu[tid] = s1;
      sh_ss[tid] = fmaxf(s2 - s1 * s1, 0.0f) + EPSF;   // sum_n coeff == 1
    }
    __syncthreads();

    // --- activation cost + log terms per o -------------------------------
    if (tid < CAPS_B) {
      const float rs = sh_rsum[tid];
      const float bu = beta_u[tid];
      float cost = 0.f, lt = 0.f;
      for (int t2 = 0; t2 < 16; ++t2) {
        const float ssv = sh_ss[tid * 16 + t2];
        cost += (bu + 0.5f * logf(ssv)) * rs;
        lt   += logf(ssv * LN2PI_F);
      }
      const float ao = 1.0f / (1.0f + expf(-lam * (beta_a[tid] - cost)));
      sh_aout[tid]    = ao;
      sh_la[tid]      = logf(ao);
      sh_logterm[tid] = -0.5f * lt;
    }
    __syncthreads();

    if (it == 2) break;

    // --- r update: ln_p + log a, softmax over o per n --------------------
    for (int n0 = 0; n0 < KKA; n0 += 16) {
      const int nn  = n0 + (tid >> 5);
      const int oo  = tid & 31;
      const float* vp2 =
          v_ws + (size_t)nn * VSTRIDE_N + (size_t)(bl * 4) * NCOLS + oo * 4;
      float q = 0.f;
#pragma unroll
      for (int t2 = 0; t2 < 16; ++t2) {
        const float vv = vp2[(t2 >> 2) * NCOLS + (t2 & 3)];
        const float d  = vv - sh_mu[oo * 16 + t2];
        q += d * d / (2.0f * sh_ss[oo * 16 + t2]);
      }
      sh_lnap[(tid >> 5) * 32 + oo] = sh_logterm[oo] - q + sh_la[oo];
      __syncthreads();

      const int row = tid >> 5;
      float mx = -1e30f;
      for (int o2 = 0; o2 < 32; ++o2)
        mx = fmaxf(mx, sh_lnap[row * 32 + o2]);
      float se = 0.f;
      for (int o2 = 0; o2 < 32; ++o2)
        se += expf(sh_lnap[row * 32 + o2] - mx);
      sh_r[nn * 32 + oo] = expf(sh_lnap[row * 32 + oo] - mx) / se;
      __syncthreads();
    }
  }

  // --- outputs -----------------------------------------------------------
  // pose_out[b, o*16+ij, l] = mu ; a_out[b, o, l]
  out_pose[((size_t)b * 512 + o * 16 + ij) * NL + l] = sh_mu[tid];
  if (tid < CAPS_B)
    out_a[((size_t)b * 32 + tid) * NL + l] = sh_aout[tid];
}

// ---------------------------------------------------------------------------
extern "C" void kernel_launch(void* const* d_in, const int* in_sizes, int n_in,
                              void* d_out, int out_size, void* d_ws,
                              size_t ws_size, hipStream_t stream) {
  const float* a_in   = (const float*)d_in[0];   // 8*32*16*16
  const float* pose   = (const float*)d_in[1];   // 8*512*16*16
  const float* W      = (const float*)d_in[2];   // 288*32*4*4
  const float* beta_u = (const float*)d_in[3];   // 32
  const float* beta_a = (const float*)d_in[4];   // 32

  float* out_a    = (float*)d_out;               // 8*32*49 = 12544
  float* out_pose = out_a + 8 * 32 * NL;         // 8*512*49 = 200704
  float* v_ws     = (float*)d_ws;                // needs 288*1568*128*4 B ~ 231 MB

  // Kernel 1: one wave32 per (row-tile, n); 8 WMMAs per wave
  convcaps_pose_wmma<<<dim3(98, KKA), dim3(32), 0, stream>>>(pose, W, v_ws);

  // Kernel 2: one 512-thread block per (b,l)
  convcaps_routing<<<dim3(NBL), dim3(512), 0, stream>>>(
      a_in, v_ws, beta_u, beta_a, out_a, out_pose);
}